// MEKER_84799834292999
// MI455X (gfx1250) — compile-verified
//
#include <hip/hip_runtime.h>

// ---------------------------------------------------------------------------
// Problem constants (from the reference)
// ---------------------------------------------------------------------------
#define MK_RANK 200
#define MK_R4   (MK_RANK / 4)   // 50 float4 per row
#define MK_TAIL (MK_R4 - 32)    // 18 tail float4s (lanes 0..17)
#define MK_NNZ  500000
#define MK_NENT 200000
#define MK_NREL 500

typedef float v4f __attribute__((ext_vector_type(4)));
typedef int   v4i __attribute__((vector_size(16)));   // matches builtin param

typedef __attribute__((address_space(1))) v4i* mk_gv4i_p;  // global
typedef __attribute__((address_space(3))) v4i* mk_lv4i_p;  // LDS

// ---------------------------------------------------------------------------
// 16B async copy global -> LDS (per-lane), tracked by ASYNCcnt.
// ---------------------------------------------------------------------------
__device__ __forceinline__ void mk_async_cp16(float* lds_dst, const float* gsrc) {
    __builtin_amdgcn_global_load_async_to_lds_b128(
        (mk_gv4i_p)(void*)gsrc,
        (mk_lv4i_p)(void*)lds_dst,
        /*imm offset*/ 0, /*cpol*/ 0);
}

// Compiler-level memory barrier: pins LDS reads above the next buffer's
// async issues so the scheduler cannot hoist overwrites past consumption.
__device__ __forceinline__ void mk_sched_fence() { asm volatile("" ::: "memory"); }

// ---------------------------------------------------------------------------
// Kernel 1: zero grad_a | grad_b (contiguous in d_out, 16B aligned)
// ---------------------------------------------------------------------------
__global__ __launch_bounds__(256) void meker_zero(float* __restrict__ p, long long n4) {
    long long i      = (long long)blockIdx.x * blockDim.x + threadIdx.x;
    long long stride = (long long)gridDim.x * blockDim.x;
    const v4f z = {0.f, 0.f, 0.f, 0.f};
    v4f* q = (v4f*)p;
    for (; i < n4; i += stride) __builtin_nontemporal_store(z, q + i);
}

// ---------------------------------------------------------------------------
// Kernel 2: forward + subj/rel gradients. One wave32 per edge; 3 rows per
// edge staged in LDS via async b128 copies, double buffered (depth 2).
// ---------------------------------------------------------------------------
__global__ __launch_bounds__(256) void meker_phase1(
    const int*   __restrict__ coo,   const float* __restrict__ vals,
    const float* __restrict__ a,     const float* __restrict__ b,
    float*       __restrict__ loss,  float*       __restrict__ grad_a,
    float*       __restrict__ grad_b, float*      __restrict__ deriv_ws)
{
    __shared__ float smem[8][2][3][MK_RANK];          // 38.4 KB / block
    const int lane   = threadIdx.x & 31;
    const int wslot  = threadIdx.x >> 5;
    const int wave   = (int)((blockIdx.x * blockDim.x + threadIdx.x) >> 5);
    const int nwaves = (int)((gridDim.x * blockDim.x) >> 5);

    float (*buf)[3][MK_RANK] = smem[wslot];

    auto issue_rows = [&](float (*dst)[MK_RANK], int s, int r, int o) {
        const float* rs = a + (size_t)s * MK_RANK;
        const float* rr = b + (size_t)r * MK_RANK;
        const float* ro = a + (size_t)o * MK_RANK;
        mk_async_cp16(&dst[0][lane * 4], rs + lane * 4);
        mk_async_cp16(&dst[1][lane * 4], rr + lane * 4);
        mk_async_cp16(&dst[2][lane * 4], ro + lane * 4);
        if (lane < MK_TAIL) {
            mk_async_cp16(&dst[0][(lane + 32) * 4], rs + (lane + 32) * 4);
            mk_async_cp16(&dst[1][(lane + 32) * 4], rr + (lane + 32) * 4);
            mk_async_cp16(&dst[2][(lane + 32) * 4], ro + (lane + 32) * 4);
        }
    };

    int e = wave;
    if (e >= MK_NNZ) return;

    int cs = coo[3 * e + 0], cr = coo[3 * e + 1], co = coo[3 * e + 2];
    issue_rows(buf[0], cs, cr, co);                   // prime the pipeline
    int pb = 0;

    for (; e < MK_NNZ; e += nwaves) {
        const int  en   = e + nwaves;
        const bool more = en < MK_NNZ;
        int ns = 0, nr = 0, no = 0;
        if (more) {
            ns = coo[3 * en + 0]; nr = coo[3 * en + 1]; no = coo[3 * en + 2];
            issue_rows(buf[pb ^ 1], ns, nr, no);      // stage next edge
            if (lane == 0 && en + nwaves < MK_NNZ)    // prefetch coo line ahead
                __builtin_prefetch(coo + 3 * (en + nwaves), 0, 0);
            __builtin_amdgcn_s_wait_asynccnt(6);      // current buffer landed
        } else {
            __builtin_amdgcn_s_wait_asynccnt(0);
        }

        const v4f* LA0 = (const v4f*)buf[pb][0];
        const v4f* LB1 = (const v4f*)buf[pb][1];
        const v4f* LA2 = (const v4f*)buf[pb][2];

        v4f x0 = LA0[lane], y0 = LB1[lane], z0 = LA2[lane];
        const bool hi = lane < MK_TAIL;
        v4f x1 = {0.f, 0.f, 0.f, 0.f}, y1 = x1, z1 = x1;
        if (hi) { x1 = LA0[lane + 32]; y1 = LB1[lane + 32]; z1 = LA2[lane + 32]; }
        mk_sched_fence();   // keep these LDS reads above next iteration's asyncs

        v4f p0 = x0 * y0 * z0;
        v4f p1 = x1 * y1 * z1;
        float part = p0.x + p0.y + p0.z + p0.w + p1.x + p1.y + p1.z + p1.w;
        #pragma unroll
        for (int m = 16; m >= 1; m >>= 1) part += __shfl_xor(part, m, 32);

        const float diff = part - vals[e];
        const float dv   = 2.0f * diff;
        if (lane == 0) {
            loss[e]     = diff * diff;
            deriv_ws[e] = dv;
        }

        // g_a = dv*b1*a2 -> grad_a[subj];  g_b = dv*a0*a2 -> grad_b[rel]
        v4f ga = dv * y0 * z0;
        v4f gb = dv * x0 * z0;
        __builtin_nontemporal_store(ga, (v4f*)(grad_a + (size_t)cs * MK_RANK) + lane);
        __builtin_nontemporal_store(gb, (v4f*)(grad_b + (size_t)cr * MK_RANK) + lane);
        if (hi) {
            v4f ga1 = dv * y1 * z1;
            v4f gb1 = dv * x1 * z1;
            __builtin_nontemporal_store(ga1, (v4f*)(grad_a + (size_t)cs * MK_RANK) + lane + 32);
            __builtin_nontemporal_store(gb1, (v4f*)(grad_b + (size_t)cr * MK_RANK) + lane + 32);
        }

        cs = ns; cr = nr; co = no; pb ^= 1;
    }
}

// ---------------------------------------------------------------------------
// Kernel 3: obj gradients (stream-ordered AFTER phase1 so the obj scatter-set
// overwrites the subj scatter-set, matching the reference's chained .set()).
// Same async double-buffered staging, 2 rows per edge.
// ---------------------------------------------------------------------------
__global__ __launch_bounds__(256) void meker_phase2(
    const int*   __restrict__ coo,
    const float* __restrict__ a, const float* __restrict__ b,
    const float* __restrict__ deriv_ws, float* __restrict__ grad_a)
{
    __shared__ float smem[8][2][2][MK_RANK];          // 25.6 KB / block
    const int lane   = threadIdx.x & 31;
    const int wslot  = threadIdx.x >> 5;
    const int wave   = (int)((blockIdx.x * blockDim.x + threadIdx.x) >> 5);
    const int nwaves = (int)((gridDim.x * blockDim.x) >> 5);

    float (*buf)[2][MK_RANK] = smem[wslot];

    auto issue_rows = [&](float (*dst)[MK_RANK], int s, int r) {
        const float* rs = a + (size_t)s * MK_RANK;
        const float* rr = b + (size_t)r * MK_RANK;
        mk_async_cp16(&dst[0][lane * 4], rs + lane * 4);
        mk_async_cp16(&dst[1][lane * 4], rr + lane * 4);
        if (lane < MK_TAIL) {
            mk_async_cp16(&dst[0][(lane + 32) * 4], rs + (lane + 32) * 4);
            mk_async_cp16(&dst[1][(lane + 32) * 4], rr + (lane + 32) * 4);
        }
    };

    int e = wave;
    if (e >= MK_NNZ) return;

    int cs = coo[3 * e + 0], cr = coo[3 * e + 1], co = coo[3 * e + 2];
    issue_rows(buf[0], cs, cr);
    int pb = 0;

    for (; e < MK_NNZ; e += nwaves) {
        const int  en   = e + nwaves;
        const bool more = en < MK_NNZ;
        int ns = 0, nr = 0, no = 0;
        if (more) {
            ns = coo[3 * en + 0]; nr = coo[3 * en + 1]; no = coo[3 * en + 2];
            issue_rows(buf[pb ^ 1], ns, nr);
            if (lane == 0 && en + nwaves < MK_NNZ)
                __builtin_prefetch(coo + 3 * (en + nwaves), 0, 0);
            __builtin_amdgcn_s_wait_asynccnt(4);
        } else {
            __builtin_amdgcn_s_wait_asynccnt(0);
        }

        const float dv = deriv_ws[e];
        const v4f* LA0 = (const v4f*)buf[pb][0];
        const v4f* LB1 = (const v4f*)buf[pb][1];

        v4f x0 = LA0[lane], y0 = LB1[lane];
        const bool hi = lane < MK_TAIL;
        v4f x1 = {0.f, 0.f, 0.f, 0.f}, y1 = x1;
        if (hi) { x1 = LA0[lane + 32]; y1 = LB1[lane + 32]; }
        mk_sched_fence();   // keep these LDS reads above next iteration's asyncs

        v4f gc = dv * x0 * y0;
        __builtin_nontemporal_store(gc, (v4f*)(grad_a + (size_t)co * MK_RANK) + lane);
        if (hi) {
            v4f gc1 = dv * x1 * y1;
            __builtin_nontemporal_store(gc1, (v4f*)(grad_a + (size_t)co * MK_RANK) + lane + 32);
        }

        cs = ns; cr = nr; co = no; pb ^= 1;
    }
}

// ---------------------------------------------------------------------------
// Host entry
// ---------------------------------------------------------------------------
extern "C" void kernel_launch(void* const* d_in, const int* in_sizes, int n_in,
                              void* d_out, int out_size, void* d_ws, size_t ws_size,
                              hipStream_t stream) {
    const int*   coo  = (const int*)d_in[0];    // [NNZ,3]
    const float* vals = (const float*)d_in[1];  // [NNZ]
    const float* a    = (const float*)d_in[2];  // [N_ENT,RANK]
    const float* b    = (const float*)d_in[3];  // [N_REL,RANK]

    float* out      = (float*)d_out;
    float* loss     = out;                                    // [NNZ]
    float* grad_a   = out + MK_NNZ;                           // [N_ENT*RANK]
    float* grad_b   = grad_a + (size_t)MK_NENT * MK_RANK;     // [N_REL*RANK]
    float* deriv_ws = (float*)d_ws;                           // [NNZ] (2 MB)

    const long long n4 = ((long long)(MK_NENT + MK_NREL) * MK_RANK) / 4;  // 10,025,000
    meker_zero<<<4096, 256, 0, stream>>>(grad_a, n4);

    // 4096 blocks * 8 waves = 32768 waves -> ~15 edges/wave (async pipeline).
    meker_phase1<<<4096, 256, 0, stream>>>(coo, vals, a, b, loss, grad_a, grad_b, deriv_ws);
    meker_phase2<<<4096, 256, 0, stream>>>(coo, a, b, deriv_ws, grad_a);
}